// Net_85315230368008
// MI455X (gfx1250) — compile-verified
//
#include <hip/hip_runtime.h>
#include <math.h>

// Problem constants (match reference)
#define B_   4
#define N_   1024
#define C_   23
#define NB_  3
#define H_   100
#define HP_  112        // H padded to 7 tiles of 16
#define NT_  7          // HP_/16
#define D1_  30
#define D2_  10
#define NSPLIT_ 32
#define BN_PER_SPLIT_ (N_ / NSPLIT_)   // 32 bnodes per wave

typedef __attribute__((ext_vector_type(2))) float v2f;
typedef __attribute__((ext_vector_type(8))) float v8f;

// e3nn CosineBasisModel bump: cos(pi/2 * u)^2 for |u|<1 else 0, u=(r-center)/step
__device__ __forceinline__ float cos_bump(float r, float center) {
    float u = (r - center) * (1.0f / 1.5f);            // step = 3.0/(3-1) = 1.5
    float c = __cosf(1.57079632679f * u);
    return (fabsf(u) < 1.0f) ? c * c : 0.0f;
}

// single-instruction relu: v_med3_f32(x, 0, +inf) == clamp to [0, inf)
__device__ __forceinline__ float relu1(float x) {
    return __builtin_amdgcn_fmed3f(x, 0.0f, __builtin_inff());
}

// ---------------------------------------------------------------------------
// Kernel 1: F2[b,n,j] = sum_c W_rb2[j,c] * features[b,n,c]  (j padded to 112)
//           biasPart[b,n] = sum_c b_rb2[c] * features[b,n,c]
// ---------------------------------------------------------------------------
__global__ void k_f2(const float* __restrict__ features,
                     const float* __restrict__ W_rb2,
                     const float* __restrict__ b_rb2,
                     float* __restrict__ F2,
                     float* __restrict__ biasPart) {
    int id = blockIdx.x * blockDim.x + threadIdx.x;
    if (id >= B_ * N_ * HP_) return;
    int j  = id % HP_;
    int bn = id / HP_;                  // flat b*N + n
    const float* f = features + (size_t)bn * C_;
    float acc = 0.0f;
    if (j < H_) {
        const float* w = W_rb2 + (size_t)j * C_;
        #pragma unroll
        for (int c = 0; c < C_; ++c) acc += w[c] * f[c];
    }
    F2[id] = acc;                        // zero in padded columns
    if (j == 0) {
        float bp = 0.0f;
        #pragma unroll
        for (int c = 0; c < C_; ++c) bp += b_rb2[c] * f[c];
        biasPart[bn] = bp;
    }
}

// ---------------------------------------------------------------------------
// Kernel 2: biasSum[b] = sum_n biasPart[b,n]   (deterministic block reduce)
// ---------------------------------------------------------------------------
__global__ void k_bias_sum(const float* __restrict__ biasPart,
                           float* __restrict__ biasSum) {
    __shared__ float s[256];
    int b = blockIdx.x;
    float acc = 0.0f;
    for (int n = threadIdx.x; n < N_; n += 256) acc += biasPart[b * N_ + n];
    s[threadIdx.x] = acc;
    __syncthreads();
    for (int o = 128; o > 0; o >>= 1) {
        if (threadIdx.x < o) s[threadIdx.x] += s[threadIdx.x + o];
        __syncthreads();
    }
    if (threadIdx.x == 0) biasSum[b] = s[0];
}

// ---------------------------------------------------------------------------
// Kernel 3: fused per-edge radial MLP via V_WMMA_F32_16X16X4_F32.
//   One wave handles 16 output nodes `a` (M dim) for one batch b and a
//   1/NSPLIT slice of bnodes. Per bnode: A = basis(16a x 4) with K=3 slot = 1.0
//   (bias folded into B's K=3 row), 7 WMMAs over H tiles, single-op relu
//   (v_med3), fma with F2[bnode, Htile] broadcast along M, accumulate;
//   lane-tree reduce at the end. Branchless A-operand build (cndmask, no exec
//   juggling in the hot loop).
// ---------------------------------------------------------------------------
__global__ void __launch_bounds__(32) k_conv(const float* __restrict__ geometry,
                                             const float* __restrict__ W_rb1,
                                             const float* __restrict__ b_rb1,
                                             const float* __restrict__ F2,
                                             float* __restrict__ partials) {
    const int lane  = threadIdx.x;      // wave32
    const int hn    = lane & 15;        // position within 16-lane half
    const int kpair = lane >> 4;        // 0: K={0,1}, 1: K={2,3}
    const int split = blockIdx.x;
    const int atile = blockIdx.y;
    const int b     = blockIdx.z;
    const int a     = atile * 16 + hn;  // my M row (output node)

    const float ax = geometry[((size_t)b * N_ + a) * 3 + 0];
    const float ay = geometry[((size_t)b * N_ + a) * 3 + 1];
    const float az = geometry[((size_t)b * N_ + a) * 3 + 2];

    // B operand tiles: W_rb1 rows K=0..2, bias row in K=3. Invariant -> registers.
    // f32 B 4x16 layout: lanes0-15 VGPR{0,1}=K{0,1}; lanes16-31 VGPR{0,1}=K{2,3}.
    v2f Bt[NT_];
    #pragma unroll
    for (int t = 0; t < NT_; ++t) {
        int col = t * 16 + hn;          // H column
        float b0 = (col < H_) ? W_rb1[(kpair * 2) * H_ + col] : 0.0f;     // K=0 or K=2
        float b1;
        if (kpair == 0) b1 = (col < H_) ? W_rb1[1 * H_ + col] : 0.0f;     // K=1
        else            b1 = (col < H_) ? b_rb1[col]          : 0.0f;     // K=3 -> bias
        Bt[t].x = b0;
        Bt[t].y = b1;
    }

    // per-lane center for A.x bump: K=0 -> 0.0, K=2 -> 3.0 (register select, uniform-free)
    const float cx = kpair ? 3.0f : 0.0f;

    float acc[8];
    #pragma unroll
    for (int r = 0; r < 8; ++r) acc[r] = 0.0f;

    const int bn0 = split * BN_PER_SPLIT_;
    for (int i = 0; i < BN_PER_SPLIT_; ++i) {
        const int bn = bn0 + i;
        // broadcast load of neighbor position (same addr all lanes -> 1 request)
        const float gx = geometry[((size_t)b * N_ + bn) * 3 + 0];
        const float gy = geometry[((size_t)b * N_ + bn) * 3 + 1];
        const float gz = geometry[((size_t)b * N_ + bn) * 3 + 2];
        const float dx = ax - gx, dy = ay - gy, dz = az - gz;
        const float r  = sqrtf(dx * dx + dy * dy + dz * dz + 1e-12f);

        // A operand: 16x4 f32, lanes0-15 hold K0,K1; lanes16-31 hold K2, K3(=1.0 bias)
        // Evaluate both bumps unconditionally, select with cndmask (no divergent branch).
        const float bump_x   = cos_bump(r, cx);      // K=0 (lower) / K=2 (upper)
        const float bump_mid = cos_bump(r, 1.5f);    // K=1 (only lower half uses it)
        v2f At;
        At.x = bump_x;
        At.y = kpair ? 1.0f : bump_mid;              // K=3 bias slot = 1.0 on upper half

        // F2 row for this bnode: each lane needs F2[bn, t*16 + hn]
        const float* f2row = F2 + ((size_t)b * N_ + bn) * HP_;
        float f2v[NT_];
        #pragma unroll
        for (int t = 0; t < NT_; ++t) f2v[t] = f2row[t * 16 + hn];

        #pragma unroll
        for (int t = 0; t < NT_; ++t) {
            v8f cz = {};
            // D = basis @ W_rb1 + b_rb1  (bias via K=3 slot)
            v8f d = __builtin_amdgcn_wmma_f32_16x16x4_f32(
                        false, At, false, Bt[t], (short)0, cz, false, false);
            #pragma unroll
            for (int rr = 0; rr < 8; ++rr)
                acc[rr] += relu1(d[rr]) * f2v[t];    // v_med3 + fma
        }
    }

    // Reduce across the 16 lanes of each half (N dim); lane0/lane16 own M=rr / M=rr+8
    #pragma unroll
    for (int rr = 0; rr < 8; ++rr) {
        float v = acc[rr];
        v += __shfl_xor(v, 1, 32);
        v += __shfl_xor(v, 2, 32);
        v += __shfl_xor(v, 4, 32);
        v += __shfl_xor(v, 8, 32);
        if (hn == 0) {
            int aout = atile * 16 + rr + 8 * kpair;
            partials[(size_t)split * (B_ * N_) + b * N_ + aout] = v;
        }
    }
}

// ---------------------------------------------------------------------------
// Kernel 4: conv[b,a] = (sum_split partials + biasSum[b]) / sqrt(N)
// ---------------------------------------------------------------------------
__global__ void k_reduce(const float* __restrict__ partials,
                         const float* __restrict__ biasSum,
                         float* __restrict__ conv) {
    int id = blockIdx.x * blockDim.x + threadIdx.x;
    if (id >= B_ * N_) return;
    int b = id / N_;
    float s = 0.0f;
    #pragma unroll
    for (int sp = 0; sp < NSPLIT_; ++sp) s += partials[(size_t)sp * (B_ * N_) + id];
    conv[id] = (s + biasSum[b]) * (1.0f / 32.0f);   // 1/sqrt(1024)
}

// ---------------------------------------------------------------------------
// Kernel 5: tiny MLP head, one block
// ---------------------------------------------------------------------------
__global__ void __launch_bounds__(128) k_head(const float* __restrict__ conv,
                                              const float* __restrict__ W1,
                                              const float* __restrict__ b1,
                                              const float* __restrict__ W2,
                                              const float* __restrict__ b2,
                                              const float* __restrict__ W3,
                                              const float* __restrict__ b3,
                                              float* __restrict__ out) {
    __shared__ float x1[B_][D1_];
    __shared__ float x2[B_][D2_];
    int tid = threadIdx.x;
    if (tid < B_ * D1_) {
        int b = tid / D1_, d = tid % D1_;
        float s = b1[d];
        for (int n = 0; n < N_; ++n) s += conv[b * N_ + n] * W1[n * D1_ + d];
        x1[b][d] = relu1(s);
    }
    __syncthreads();
    if (tid < B_ * D2_) {
        int b = tid / D2_, d = tid % D2_;
        float s = b2[d];
        #pragma unroll
        for (int k = 0; k < D1_; ++k) s += x1[b][k] * W2[k * D2_ + d];
        x2[b][d] = relu1(s);
    }
    __syncthreads();
    if (tid < B_) {
        float s = b3[0];
        #pragma unroll
        for (int k = 0; k < D2_; ++k) s += x2[tid][k] * W3[k];
        out[tid] = s;
    }
}

// ---------------------------------------------------------------------------
extern "C" void kernel_launch(void* const* d_in, const int* in_sizes, int n_in,
                              void* d_out, int out_size, void* d_ws, size_t ws_size,
                              hipStream_t stream) {
    const float* features = (const float*)d_in[0];
    const float* geometry = (const float*)d_in[1];
    const float* W_rb1    = (const float*)d_in[2];
    const float* b_rb1    = (const float*)d_in[3];
    const float* W_rb2    = (const float*)d_in[4];
    const float* b_rb2    = (const float*)d_in[5];
    const float* W_fc1    = (const float*)d_in[6];
    const float* b_fc1    = (const float*)d_in[7];
    const float* W_fc2    = (const float*)d_in[8];
    const float* b_fc2    = (const float*)d_in[9];
    const float* W_fc3    = (const float*)d_in[10];
    const float* b_fc3    = (const float*)d_in[11];
    float* out = (float*)d_out;

    // Workspace layout (floats): ~2.4 MB total, fully written before read.
    float* ws       = (float*)d_ws;
    float* F2       = ws;                                   // B*N*HP
    float* biasPart = F2 + (size_t)B_ * N_ * HP_;           // B*N
    float* biasSum  = biasPart + (size_t)B_ * N_;           // 16 (padded)
    float* partials = biasSum + 16;                         // NSPLIT*B*N
    float* conv     = partials + (size_t)NSPLIT_ * B_ * N_; // B*N

    k_f2<<<(B_ * N_ * HP_ + 255) / 256, 256, 0, stream>>>(
        features, W_rb2, b_rb2, F2, biasPart);

    k_bias_sum<<<B_, 256, 0, stream>>>(biasPart, biasSum);

    dim3 g3(NSPLIT_, N_ / 16, B_);
    k_conv<<<g3, 32, 0, stream>>>(geometry, W_rb1, b_rb1, F2, partials);

    k_reduce<<<(B_ * N_ + 255) / 256, 256, 0, stream>>>(partials, biasSum, conv);

    k_head<<<1, 128, 0, stream>>>(conv, W_fc1, b_fc1, W_fc2, b_fc2, W_fc3, b_fc3, out);
}